// MLSTMCell_14078902796638
// MI455X (gfx1250) — compile-verified
//
#include <hip/hip_runtime.h>
#include <math.h>

typedef float v2f __attribute__((ext_vector_type(2)));
typedef float v4f __attribute__((ext_vector_type(4)));
typedef float v8f __attribute__((ext_vector_type(8)));

constexpr int HIDDEN = 4096;
constexpr int HEADS  = 8;
constexpr int HEAD   = 512;
constexpr int BATCH  = 64;
constexpr float CAPV   = 30.0f;
constexpr float EPSV   = 1e-6f;
constexpr float GN_EPS = 1e-5f;
constexpr float KSCALE = 0.04419417382415922f; // 1/sqrt(512)
constexpr int KTOT    = 3 * HIDDEN;            // 12288
constexpr int KCHUNKS = 12;                    // 12 x 1024; chunks never straddle q/k/v
constexpr int KCHUNK  = KTOT / KCHUNKS;        // 1024

// ---------------------------------------------------------------------------
// Kernel 1: gate GEMM  raw[64][16] = qkv[64x12288] @ W^T[12288x16]
// cols 0-7 = igate heads, cols 8-15 = fgate heads.
// V_WMMA_F32_16X16X4_F32; 4 blocks (M tiles of 16), 12 waves/block = K chunks.
// Each 1024-wide K chunk is wave-uniformly inside exactly one of q/k/v, so the
// source base pointer is an SGPR constant and the inner loop is branch-free.
// Deterministic: partials reduced through LDS, no float atomics.
// ---------------------------------------------------------------------------
__global__ __launch_bounds__(384) void gate_gemm(
    const float* __restrict__ q, const float* __restrict__ k,
    const float* __restrict__ v,
    const float* __restrict__ igw, const float* __restrict__ fgw,
    float* __restrict__ raw) {
  __shared__ float sAcc[KCHUNKS * 256];
  const int mt   = blockIdx.x;          // 0..3
  const int wave = threadIdx.x >> 5;    // 0..11 : K chunk
  const int lane = threadIdx.x & 31;
  const int n    = lane & 15;           // B column (and A row-within-tile)
  const int khi  = (lane >> 4) << 1;    // lanes 0-15 -> K{0,1}, 16-31 -> K{2,3}
  const int brow = mt * 16 + n;         // batch row (A: M = lane&15)

  // region is wave-uniform: chunk w covers [w*1024, w*1024+1024)
  const int region = wave >> 2;                     // 0=q, 1=k, 2=v
  const float* base = (region == 0) ? q : (region == 1) ? k : v;
  const float* arow = base + (size_t)brow * HIDDEN + (wave & 3) * KCHUNK;
  const float* wcol = ((n < 8) ? (igw + (size_t)n * KTOT)
                               : (fgw + (size_t)(n - 8) * KTOT)) + wave * KCHUNK;

  const v2f* a2 = (const v2f*)arow;   // 8B aligned: offsets even
  const v2f* w2 = (const v2f*)wcol;

  v8f c0 = {0.f, 0.f, 0.f, 0.f, 0.f, 0.f, 0.f, 0.f};
  v8f c1 = {0.f, 0.f, 0.f, 0.f, 0.f, 0.f, 0.f, 0.f};
#pragma unroll 4
  for (int kk = 0; kk < KCHUNK; kk += 8) {
    const int j0 = (kk + khi) >> 1;       // v2f index for K-pair {0,1} or {2,3}
    const int j1 = j0 + 2;                // next group of 4
    v2f a0 = a2[j0];
    v2f b0 = w2[j0];
    v2f a1 = a2[j1];
    v2f b1 = w2[j1];
    c0 = __builtin_amdgcn_wmma_f32_16x16x4_f32(false, a0, false, b0,
                                               (short)0, c0, false, false);
    c1 = __builtin_amdgcn_wmma_f32_16x16x4_f32(false, a1, false, b1,
                                               (short)0, c1, false, false);
  }
  const v8f c = c0 + c1;
  // D layout: VGPR r -> M = r (lanes 0-15) or M = 8+r (lanes 16-31), N = lane&15
  const int hiOff = (lane >> 4) * 8;
#pragma unroll
  for (int r = 0; r < 8; ++r)
    sAcc[wave * 256 + (r + hiOff) * 16 + n] = c[r];
  __syncthreads();
  if (threadIdx.x < 256) {
    float s = 0.f;
#pragma unroll
    for (int w = 0; w < KCHUNKS; ++w) s += sAcc[w * 256 + threadIdx.x];
    raw[mt * 256 + threadIdx.x] = s;   // raw[(mt*16+m)*16 + n]
  }
}

// ---------------------------------------------------------------------------
// Kernel 2: scalar gate chain per (b,h): cap-tanh, log-sigmoid, max renorm.
// Emits {i_gate, f_gate, exp(-max_new)} to ws and max_new to d_out.
// ---------------------------------------------------------------------------
__global__ __launch_bounds__(512) void gate_finalize(
    const float* __restrict__ raw, const float* __restrict__ igb,
    const float* __restrict__ fgb, const float* __restrict__ max_state,
    float* __restrict__ gates, float* __restrict__ max_new_out) {
  const int t = threadIdx.x;           // t = b*8 + h
  if (t >= BATCH * HEADS) return;
  const int b = t >> 3, h = t & 7;
  float ig = raw[b * 16 + h]     + igb[h];
  float fg = raw[b * 16 + 8 + h] + fgb[h];
  ig = CAPV * tanhf(ig / CAPV);
  fg = CAPV * tanhf(fg / CAPV);
  const float log_f = fminf(fg, 0.f) - log1pf(expf(-fabsf(fg)));
  const float m_old = max_state[t];
  const float m_new = fmaxf(ig, m_old + log_f);
  gates[t * 4 + 0] = expf(ig - m_new);            // i_gate
  gates[t * 4 + 1] = expf(log_f + m_old - m_new); // f_gate
  gates[t * 4 + 2] = expf(-m_new);                // denom floor
  max_new_out[t]   = m_new;
}

__device__ __forceinline__ float block_reduce_sum(float val, float* sRed, int tid) {
  sRed[tid] = val;
  __syncthreads();
  for (int s = 256; s > 0; s >>= 1) {
    if (tid < s) sRed[tid] += sRed[tid + s];
    __syncthreads();
  }
  const float r = sRed[0];
  __syncthreads();
  return r;
}

// ---------------------------------------------------------------------------
// Kernel 3: HBM-bound stream. One block per (b,h). 512 threads.
// Streams 512x512 cell tile once (NT b128 load) -> f*C + i*k(x)v -> NT b128
// store, fusing numerator = q . C_new. Norm/denom computed first; group-norm locally after.
// ---------------------------------------------------------------------------
__global__ __launch_bounds__(512) void mlstm_stream(
    const float* __restrict__ q, const float* __restrict__ k,
    const float* __restrict__ v, const float* __restrict__ cell_state,
    const float* __restrict__ norm_state, const float* __restrict__ gnw,
    const float* __restrict__ gnb, const float* __restrict__ gates,
    float* __restrict__ out, float* __restrict__ cell_new,
    float* __restrict__ norm_new) {
  __shared__ float sQ[HEAD];
  __shared__ float sK[HEAD];
  __shared__ float sP[4 * HEAD];
  __shared__ float sRed[512];

  const int bh  = blockIdx.x;
  const int b   = bh >> 3;
  const int h   = bh & 7;
  const int tid = threadIdx.x;

  const float i_g  = gates[bh * 4 + 0];
  const float f_g  = gates[bh * 4 + 1];
  const float emax = gates[bh * 4 + 2];

  const float* qrow = q + (size_t)b * HIDDEN + h * HEAD;
  const float* krow = k + (size_t)b * HIDDEN + h * HEAD;
  const float* vrow = v + (size_t)b * HIDDEN + h * HEAD;

  const float qd = qrow[tid];
  const float kd = krow[tid] * KSCALE;
  sQ[tid] = qd;
  sK[tid] = kd;

  // norm_new + qn_dot
  const float nn = f_g * norm_state[(size_t)bh * HEAD + tid] + i_g * kd;
  norm_new[(size_t)bh * HEAD + tid] = nn;
  const float qn_dot = block_reduce_sum(qd * nn, sRed, tid);  // also fences sQ/sK
  const float denom  = fmaxf(fabsf(qn_dot), emax) + EPSV;

  // main stream: rg = row group (0..3), c = float4 column group (0..127)
  const int rg = tid >> 7;
  const int c  = tid & 127;
  const v4f  v4 = ((const v4f*)vrow)[c];
  const v4f* cin  = (const v4f*)(cell_state + (size_t)bh * HEAD * HEAD);
  v4f*       cout = (v4f*)(cell_new + (size_t)bh * HEAD * HEAD);

  v4f acc = {0.f, 0.f, 0.f, 0.f};
#pragma unroll 8
  for (int d0 = 0; d0 < HEAD; d0 += 4) {
    const int d   = d0 + rg;
    const float qv = sQ[d];           // wave-uniform LDS broadcast
    const float ik = i_g * sK[d];
    const int idx  = d * (HEAD / 4) + c;
    v4f cOld = __builtin_nontemporal_load(cin + idx);
    v4f cNew;
    cNew.x = f_g * cOld.x + ik * v4.x;
    cNew.y = f_g * cOld.y + ik * v4.y;
    cNew.z = f_g * cOld.z + ik * v4.z;
    cNew.w = f_g * cOld.w + ik * v4.w;
    __builtin_nontemporal_store(cNew, cout + idx);
    acc.x += qv * cNew.x;
    acc.y += qv * cNew.y;
    acc.z += qv * cNew.z;
    acc.w += qv * cNew.w;
  }

  // reduce numerator partials over the 4 row groups
  ((v4f*)sP)[rg * 128 + c] = acc;
  __syncthreads();
  const float num = sP[tid] + sP[HEAD + tid] + sP[2 * HEAD + tid] + sP[3 * HEAD + tid];
  const float o   = num / denom;

  // group norm over this head's 512 channels
  const float mean = block_reduce_sum(o, sRed, tid) * (1.0f / HEAD);
  const float diff = o - mean;
  const float var  = block_reduce_sum(diff * diff, sRed, tid) * (1.0f / HEAD);
  const float xn   = diff * rsqrtf(var + GN_EPS);
  const int ch = h * HEAD + tid;
  out[(size_t)b * HIDDEN + ch] = xn * gnw[ch] + gnb[ch];
}

// ---------------------------------------------------------------------------
extern "C" void kernel_launch(void* const* d_in, const int* in_sizes, int n_in,
                              void* d_out, int out_size, void* d_ws, size_t ws_size,
                              hipStream_t stream) {
  (void)in_sizes; (void)n_in; (void)out_size; (void)ws_size;
  const float* q    = (const float*)d_in[0];
  const float* k    = (const float*)d_in[1];
  const float* v    = (const float*)d_in[2];
  const float* cell = (const float*)d_in[3];
  const float* norm = (const float*)d_in[4];
  const float* mx   = (const float*)d_in[5];
  const float* igw  = (const float*)d_in[6];
  const float* igb  = (const float*)d_in[7];
  const float* fgw  = (const float*)d_in[8];
  const float* fgb  = (const float*)d_in[9];
  const float* gnw  = (const float*)d_in[10];
  const float* gnb  = (const float*)d_in[11];

  float* out      = (float*)d_out;                                   // [64*4096]
  float* cell_new = out + (size_t)BATCH * HIDDEN;                    // [64*8*512*512]
  float* norm_new = cell_new + (size_t)BATCH * HEADS * HEAD * HEAD;  // [64*8*512]
  float* max_new  = norm_new + (size_t)BATCH * HEADS * HEAD;         // [64*8]

  float* raw   = (float*)d_ws;   // [64][16]
  float* gates = raw + 1024;     // [64*8][4]

  gate_gemm<<<dim3(4), 384, 0, stream>>>(q, k, v, igw, fgw, raw);
  gate_finalize<<<1, 512, 0, stream>>>(raw, igb, fgb, mx, gates, max_new);
  mlstm_stream<<<BATCH * HEADS, 512, 0, stream>>>(q, k, v, cell, norm, gnw, gnb,
                                                  gates, out, cell_new, norm_new);
}